// MinGRU_71777493450934
// MI455X (gfx1250) — compile-verified
//
#include <hip/hip_runtime.h>

// MinGRU forward for MI455X (gfx1250, wave32).
// - GEMM uses V_WMMA_F32_16X16X4_F32 (fp32-native WMMA), W staged in LDS.
// - Recurrence done as a 3-pass chunked linear scan (exactly equivalent to the
//   reference's log-space cumlogsumexp formulation since all terms are > 0).

typedef float v2f __attribute__((ext_vector_type(2)));
typedef float v8f __attribute__((ext_vector_type(8)));

namespace {
constexpr int BB   = 8;
constexpr int S    = 8192;
constexpr int DH   = 256;
constexpr int K    = 256;        // DIN == DH == 256 for both layers
constexpr int N2   = 512;        // 2*DH
constexpr int M    = BB * S;     // 65536 rows
constexpr int BLK_M = 128;       // 8 waves * 16 rows
constexpr int BLK_N = 32;        // 2 wmma tiles per wave
constexpr int LDSW  = 260;       // padded LDS row stride (floats): bank stride 4 -> conflict-free
constexpr int CH   = 64;         // scan chunks per sequence
constexpr int CL   = S / CH;     // 128 steps per chunk
}

// -------------------- GEMM: GH[m][n] = sum_k X[m][k]*W[n][k] + bias[n] ------
__global__ __launch_bounds__(256) void mingru_gemm(const float* __restrict__ X,
                                                   const float* __restrict__ W,
                                                   const float* __restrict__ bias,
                                                   float* __restrict__ GH) {
  __shared__ float sW[BLK_N * LDSW];   // 32 rows x 256 K, padded (33 KB)

  const int tileM = blockIdx.x * BLK_M;
  const int tileN = blockIdx.y * BLK_N;
  const int tid   = threadIdx.x;
  const int wave  = tid >> 5;
  const int lane  = tid & 31;

  // Cooperative load of W[tileN .. tileN+32) rows into LDS (float4 per op).
  for (int e = tid; e < BLK_N * (K / 4); e += 256) {   // 2048 float4 elements
    const int n  = e >> 6;           // e / (K/4)
    const int k4 = e & 63;
    const float4 w = *reinterpret_cast<const float4*>(
        W + (size_t)(tileN + n) * K + (size_t)k4 * 4);
    *reinterpret_cast<float4*>(&sW[n * LDSW + k4 * 4]) = w;  // 260*4 B = 16*65 -> 16B aligned
  }
  __syncthreads();

  // fp32 16x16x4 A-fragment layout: lanes 0-15 hold M=lane, K={0,1};
  // lanes 16-31 hold M=lane-16, K={2,3}.
  const int rowBase = tileM + wave * 16;
  const int mrow    = rowBase + (lane & 15);
  const int kh      = (lane >> 4) * 2;          // 0 or 2
  const float* aPtr = X + (size_t)mrow * K + kh;
  const int nlo     = lane & 15;

  v8f acc[2] = {};  // two 16x16 fp32 accumulators (BLK_N = 32)

  for (int k0 = 0; k0 < K; k0 += 4) {
    const v2f a = *reinterpret_cast<const v2f*>(aPtr + k0);
#pragma unroll
    for (int j = 0; j < 2; ++j) {
      // B-fragment (4x16): lanes 0-15 = (K=k0+kh, N=lane), lanes 16-31 = (K=k0+2+.., N=lane-16)
      const v2f b = *reinterpret_cast<const v2f*>(
          &sW[(j * 16 + nlo) * LDSW + kh + k0]);
      acc[j] = __builtin_amdgcn_wmma_f32_16x16x4_f32(
          /*neg_a=*/false, a, /*neg_b=*/false, b,
          /*c_mod=*/(short)0, acc[j], /*reuse_a=*/false, /*reuse_b=*/false);
    }
  }

  // C/D layout: VGPR r -> lanes 0-15: (M=r, N=lane); lanes 16-31: (M=8+r, N=lane-16)
  const int colLane = lane & 15;
  const int rowOff  = (lane < 16) ? 0 : 8;
#pragma unroll
  for (int j = 0; j < 2; ++j) {
    const int col  = tileN + j * 16 + colLane;
    const float bv = bias[col];
#pragma unroll
    for (int r = 0; r < 8; ++r) {
      const int row = rowBase + rowOff + r;
      GH[(size_t)row * N2 + col] = acc[j][r] + bv;
    }
  }
}

// -------------------- scan epilogue math ------------------------------------
__device__ __forceinline__ void gru_step_terms(float gate, float hid,
                                               float& z, float& gv) {
  z  = 1.0f / (1.0f + __expf(-gate));                               // sigmoid
  gv = (hid >= 0.0f) ? (hid + 0.5f) : 1.0f / (1.0f + __expf(-hid)); // g(x)
}

// Pass 1: per-chunk aggregates. For chunk of length CL:
//   A = prod (1-z),  V = result of h' = (1-z) h' + z g  starting from h'=0.
__global__ __launch_bounds__(256) void mingru_scan_chunk(const float* __restrict__ GH,
                                                         float* __restrict__ Acoef,
                                                         float* __restrict__ Vval) {
  const int blk = blockIdx.x;            // b*CH + c
  const int b   = blk / CH;
  const int c   = blk % CH;
  const int d   = threadIdx.x;           // channel within DH
  const size_t base = ((size_t)(b * S + c * CL)) * N2 + d;

  float A = 1.0f, V = 0.0f;
#pragma unroll 4
  for (int t = 0; t < CL; ++t) {
    const size_t g = base + (size_t)t * N2;
    __builtin_prefetch(&GH[g + (size_t)16 * N2], 0, 0);
    float z, gv;
    gru_step_terms(GH[g], GH[g + DH], z, gv);
    const float cc = 1.0f - z;
    A *= cc;
    V = fmaf(cc, V, z * gv);
  }
  Acoef[(size_t)blk * DH + d] = A;
  Vval [(size_t)blk * DH + d] = V;
}

// Pass 2: tiny cross-chunk sequential fixup; record h at each chunk start.
__global__ __launch_bounds__(256) void mingru_scan_fixup(const float* __restrict__ Acoef,
                                                         const float* __restrict__ Vval,
                                                         float* __restrict__ Hstart) {
  const int b = blockIdx.x;
  const int d = threadIdx.x;
  float h = 0.5f;                        // h0 = g(0) = 0.5
  for (int c = 0; c < CH; ++c) {
    const size_t i = ((size_t)b * CH + c) * DH + d;
    Hstart[i] = h;
    h = fmaf(Acoef[i], h, Vval[i]);
  }
}

// Pass 3: re-scan each chunk from its correct start state, emit outputs.
__global__ __launch_bounds__(256) void mingru_scan_apply(const float* __restrict__ GH,
                                                         const float* __restrict__ Hstart,
                                                         float* __restrict__ Out,
                                                         float* __restrict__ Hlast) {
  const int blk = blockIdx.x;
  const int b   = blk / CH;
  const int c   = blk % CH;
  const int d   = threadIdx.x;
  const size_t base  = ((size_t)(b * S + c * CL)) * N2 + d;
  const size_t obase = ((size_t)(b * S + c * CL)) * DH + d;

  float h = Hstart[(size_t)blk * DH + d];
#pragma unroll 4
  for (int t = 0; t < CL; ++t) {
    const size_t g = base + (size_t)t * N2;
    __builtin_prefetch(&GH[g + (size_t)16 * N2], 0, 0);
    float z, gv;
    gru_step_terms(GH[g], GH[g + DH], z, gv);
    h = fmaf(z, gv - h, h);              // (1-z)h + z*gv
    Out[obase + (size_t)t * DH] = h;
  }
  if (c == CH - 1) Hlast[b * DH + d] = h;
}

// -------------------- launch -------------------------------------------------
extern "C" void kernel_launch(void* const* d_in, const int* in_sizes, int n_in,
                              void* d_out, int out_size, void* d_ws, size_t ws_size,
                              hipStream_t stream) {
  const float* x  = (const float*)d_in[0];
  const float* W0 = (const float*)d_in[1];
  const float* b0 = (const float*)d_in[2];
  const float* W1 = (const float*)d_in[3];
  const float* b1 = (const float*)d_in[4];
  float* out = (float*)d_out;

  // Workspace: gh (M*N2 fp32 = 128 MB), then small scan scratch (1.5 MB).
  float* gh  = (float*)d_ws;
  float* Ac  = gh + (size_t)M * N2;
  float* Vv  = Ac + (size_t)BB * CH * DH;
  float* Hst = Vv + (size_t)BB * CH * DH;

  // out0 lives in d_out and is later overwritten by out1 (layer-1 GEMM fully
  // consumes it first; stream order guarantees this).
  float* out0  = out;
  float* hlast = out + (size_t)M * DH;   // h[0] at +0, h[1] at +BB*DH

  const dim3 gGrid(M / BLK_M, N2 / BLK_N);   // (512, 16)

  // ---- Layer 0 ----
  mingru_gemm<<<gGrid, 256, 0, stream>>>(x, W0, b0, gh);
  mingru_scan_chunk<<<BB * CH, 256, 0, stream>>>(gh, Ac, Vv);
  mingru_scan_fixup<<<BB, 256, 0, stream>>>(Ac, Vv, Hst);
  mingru_scan_apply<<<BB * CH, 256, 0, stream>>>(gh, Hst, out0, hlast);

  // ---- Layer 1 ----
  mingru_gemm<<<gGrid, 256, 0, stream>>>(out0, W1, b1, gh);
  mingru_scan_chunk<<<BB * CH, 256, 0, stream>>>(gh, Ac, Vv);
  mingru_scan_fixup<<<BB, 256, 0, stream>>>(Ac, Vv, Hst);
  mingru_scan_apply<<<BB * CH, 256, 0, stream>>>(gh, Hst, out, hlast + BB * DH);
}